// EquivarLayer_72146860638533
// MI455X (gfx1250) — compile-verified
//
#include <hip/hip_runtime.h>

typedef float v2f __attribute__((ext_vector_type(2)));
typedef float v4f __attribute__((ext_vector_type(4)));
typedef float v8f __attribute__((ext_vector_type(8)));

#define N_ATOMS 10000
#define N_PAIRS 250000
#define N_PROP  128
#define N_OUT   128
#define ROWS    (N_ATOMS * 3)            // 30000 rows for the GEMM

// d_out layout (flat f32, reference return order):
//   px_new : N_ATOMS*3*N_OUT            = 3,840,000
//   ix     : N_PAIRS*3*N_PROP           = 96,000,000
//   dotted : N_ATOMS*N_OUT              = 1,280,000
#define OFF_PXNEW 0
#define OFF_IX    ((size_t)N_ATOMS * 3 * N_OUT)
#define OFF_DOT   (OFF_IX + (size_t)N_PAIRS * 3 * N_PROP)

// ---------------------------------------------------------------------------
// Kernel 1: zero the segment-sum accumulator in workspace (15.36 MB, L2-hot)
// ---------------------------------------------------------------------------
__global__ void zero_acc_kernel(float* __restrict__ acc, int n) {
    int i = blockIdx.x * blockDim.x + threadIdx.x;
    if (i < n) acc[i] = 0.0f;
}

// ---------------------------------------------------------------------------
// Kernel 2: per-pair fused compute (bandwidth-bound phase, ~900 MB traffic).
//   ix[p,x,:] = (px[j,x,:] + diff[p,x]) * i1[p,:]
//   acc[i,x,:] += ix[p,x,:]            (hardware global_atomic_add_f32)
// Streaming data (i1 in, ix out) uses non-temporal cache policy so the hot
// working set (px gather table, atomic accumulator) stays resident in L2.
// block = (32, 3, 2): lane handles 4 props (float4), y = x-dim, z = pair slot
// ---------------------------------------------------------------------------
__global__ __launch_bounds__(192)
void pair_kernel(const int*   __restrict__ ind2,
                 const float* __restrict__ px,
                 const float* __restrict__ i1,
                 const float* __restrict__ diff,
                 float*       __restrict__ ix_out,
                 float*       __restrict__ acc) {
    const int pair = blockIdx.x * 2 + threadIdx.z;     // N_PAIRS even, no tail
    const int x    = threadIdx.y;                      // 0..2
    const int p0   = threadIdx.x * 4;                  // prop base 0..124

    const int i = ind2[pair * 2 + 0];
    const int j = ind2[pair * 2 + 1];
    const float d = diff[pair * 3 + x];

    // i1 is streaming (never re-read): non-temporal load -> TH_NT.
    const v4f s  = __builtin_nontemporal_load(
        (const v4f*)(i1 + (size_t)pair * N_PROP + p0));
    // px is a 15 MB gather table reused across all 250k pairs: keep RT (L2-hot).
    const v4f pj = *(const v4f*)(px + ((size_t)j * 3 + x) * N_PROP + p0);

    v4f v;
    v.x = (pj.x + d) * s.x;
    v.y = (pj.y + d) * s.y;
    v.z = (pj.z + d) * s.z;
    v.w = (pj.w + d) * s.w;

    // ix (384 MB) is write-once streaming output: non-temporal store.
    __builtin_nontemporal_store(
        v, (v4f*)(ix_out + ((size_t)pair * 3 + x) * N_PROP + p0));

    float* dst = acc + ((size_t)i * 3 + x) * N_PROP + p0;
    unsafeAtomicAdd(dst + 0, v.x);
    unsafeAtomicAdd(dst + 1, v.y);
    unsafeAtomicAdd(dst + 2, v.z);
    unsafeAtomicAdd(dst + 3, v.w);
}

// ---------------------------------------------------------------------------
// Kernel 3: px_new = acc(30000x128) @ w_pp(128x128) via V_WMMA_F32_16X16X4_F32
// One block per M-tile (16 rows); 8 waves, one 16-col N-tile per wave.
// w_pp staged once per block in 64 KB LDS (WGP has 320 KB).
// Lane mapping (ISA 7.12.2, f32):
//   A 16x4 : a[v] = A[lane&15][kbase + v + 2*(lane>>4)]  -> contiguous float2
//   B 4x16 : b[v] = B[kbase + v + 2*(lane>>4)][lane&15]
//   D 16x16: d[r] = D[r + 8*(lane>>4)][lane&15]
// ---------------------------------------------------------------------------
__global__ __launch_bounds__(256)
void gemm_wmma_kernel(const float* __restrict__ acc,
                      const float* __restrict__ w,
                      float*       __restrict__ pxnew) {
    __shared__ float w_lds[N_PROP * N_OUT];            // 64 KB

    for (int idx = threadIdx.x; idx < (N_PROP * N_OUT) / 4; idx += 256) {
        *(v4f*)(w_lds + idx * 4) = *(const v4f*)(w + idx * 4);
    }
    __syncthreads();

    const int wave  = threadIdx.x >> 5;                // N-tile 0..7
    const int lane  = threadIdx.x & 31;
    const int n     = lane & 15;
    const int h     = lane >> 4;
    const int mtile = blockIdx.x;                      // 0..1874

    const float* arow = acc + ((size_t)mtile * 16 + n) * N_PROP;  // row m = lane&15
    const int    ncol = wave * 16 + n;

    v8f c = {0.f, 0.f, 0.f, 0.f, 0.f, 0.f, 0.f, 0.f};

    #pragma unroll 4
    for (int kk = 0; kk < N_PROP / 4; ++kk) {
        const int kbase = kk * 4 + 2 * h;
        v2f a = *(const v2f*)(arow + kbase);           // 8B-aligned (kbase even)
        v2f b;
        b.x = w_lds[(kbase + 0) * N_OUT + ncol];
        b.y = w_lds[(kbase + 1) * N_OUT + ncol];
        c = __builtin_amdgcn_wmma_f32_16x16x4_f32(
                /*neg_a=*/false, a, /*neg_b=*/false, b,
                /*c_mod=*/(short)0, c, /*reuse_a=*/false, /*reuse_b=*/false);
    }

    float* out = pxnew + ((size_t)mtile * 16 + 8 * h) * N_OUT + ncol;
    #pragma unroll
    for (int r = 0; r < 8; ++r) {
        out[(size_t)r * N_OUT] = c[r];
    }
}

// ---------------------------------------------------------------------------
// Kernel 4: dotted_px[a,q] = sum_x px_new[a,x,q]^2
// ---------------------------------------------------------------------------
__global__ void dotted_kernel(const float* __restrict__ pxnew,
                              float*       __restrict__ dot) {
    int t = blockIdx.x * blockDim.x + threadIdx.x;
    if (t >= N_ATOMS * N_OUT) return;
    int a = t >> 7;
    int q = t & (N_OUT - 1);
    const float* base = pxnew + (size_t)a * 3 * N_OUT + q;
    float v0 = base[0];
    float v1 = base[N_OUT];
    float v2 = base[2 * N_OUT];
    __builtin_nontemporal_store(v0 * v0 + v1 * v1 + v2 * v2, dot + t);
}

// ---------------------------------------------------------------------------
extern "C" void kernel_launch(void* const* d_in, const int* in_sizes, int n_in,
                              void* d_out, int out_size, void* d_ws, size_t ws_size,
                              hipStream_t stream) {
    const int*   ind2 = (const int*)  d_in[0];   // (250000, 2) int32
    const float* px   = (const float*)d_in[1];   // (10000, 3, 128)
    const float* i1   = (const float*)d_in[2];   // (250000, 128)
    const float* diff = (const float*)d_in[3];   // (250000, 3)
    const float* w    = (const float*)d_in[4];   // (128, 128)

    float* out   = (float*)d_out;
    float* pxnew = out + OFF_PXNEW;
    float* ixout = out + OFF_IX;
    float* dot   = out + OFF_DOT;
    float* acc   = (float*)d_ws;                 // 30000*128 f32 = 15.36 MB

    const int accN = ROWS * N_PROP;
    zero_acc_kernel<<<(accN + 255) / 256, 256, 0, stream>>>(acc, accN);

    pair_kernel<<<N_PAIRS / 2, dim3(32, 3, 2), 0, stream>>>(
        ind2, px, i1, diff, ixout, acc);

    gemm_wmma_kernel<<<ROWS / 16, 256, 0, stream>>>(acc, w, pxnew);

    dotted_kernel<<<(N_ATOMS * N_OUT + 255) / 256, 256, 0, stream>>>(pxnew, dot);
}